// HitchNet_39762807226538
// MI455X (gfx1250) — compile-verified
//
#include <hip/hip_runtime.h>
#include <math.h>

// =====================================================================
// HitchNet forward for MI455X (gfx1250), wave32, WMMA f32 path.
//
// All large GEMMs (encoder qkv/proj/FF, point MLP layer2, GAT x@W^T,
// per-batch x@x^T distance GEMM, fusion K/V projections) run through
// V_WMMA_F32_16X16X4_F32. One wave owns a 16x64 output strip (4 tiles,
// A fragment reused 4x), K stepped by 4, fp32 accumulate (needed: kNN
// distances rely on cancellation in sq_i + sq_j - 2*dot).
// All call sites have M%16==0, N%64==0, K%4==0, even lda/ldw ->
// branch-free inner loop of aligned b64 loads + 4 wmmas.
// =====================================================================

typedef __attribute__((ext_vector_type(2))) float v2f;
typedef __attribute__((ext_vector_type(8))) float v8f;

#define HN_NT 4   // N-tiles per wave (16x64 strip)

__global__ __launch_bounds__(256) void hn_gemm_wmma_fast(
    const float* __restrict__ A, const float* __restrict__ W,
    const float* __restrict__ bias, float* __restrict__ C,
    int M, int N, int K, int lda, int ldw, int ldc, int relu)
{
    int gwave   = (blockIdx.x * blockDim.x + threadIdx.x) >> 5;
    int lane    = threadIdx.x & 31;
    int nstrips = (N >> 4) / HN_NT;        // caller guarantees divisibility
    int tM = (gwave / nstrips) << 4;
    int tN = (gwave % nstrips) * (HN_NT * 16);
    if (tM >= M) return;

    int l15   = lane & 15;
    int khalf = (lane >> 4) << 1;          // 0 or 2: this half-wave's K pair

    // Lane layout per CDNA5 ISA 7.12.2 (f32 16x16x4):
    //   A: lane l -> A[tM+(l&15)][k0 + 2*(l>>4) + {0,1}]
    //   B: lane l -> W[n      ][k0 + 2*(l>>4) + {0,1}]   (B = W^T)
    const float* aptr = A + (size_t)(tM + l15) * lda + khalf;
    const float* bptr0 = W + (size_t)(tN +  0 + l15) * ldw + khalf;
    const float* bptr1 = W + (size_t)(tN + 16 + l15) * ldw + khalf;
    const float* bptr2 = W + (size_t)(tN + 32 + l15) * ldw + khalf;
    const float* bptr3 = W + (size_t)(tN + 48 + l15) * ldw + khalf;

    v8f acc0 = {}, acc1 = {}, acc2 = {}, acc3 = {};

    #pragma unroll 2
    for (int k0 = 0; k0 < K; k0 += 4) {
        v2f a  = *(const v2f*)(aptr  + k0);
        v2f b0 = *(const v2f*)(bptr0 + k0);
        v2f b1 = *(const v2f*)(bptr1 + k0);
        v2f b2 = *(const v2f*)(bptr2 + k0);
        v2f b3 = *(const v2f*)(bptr3 + k0);
        acc0 = __builtin_amdgcn_wmma_f32_16x16x4_f32(false, a, false, b0, (short)0, acc0, false, false);
        acc1 = __builtin_amdgcn_wmma_f32_16x16x4_f32(false, a, false, b1, (short)0, acc1, false, false);
        acc2 = __builtin_amdgcn_wmma_f32_16x16x4_f32(false, a, false, b2, (short)0, acc2, false, false);
        acc3 = __builtin_amdgcn_wmma_f32_16x16x4_f32(false, a, false, b3, (short)0, acc3, false, false);
    }

    // C/D layout: VGPR r, lanes0-15 -> row tM+r, lanes16-31 -> row tM+r+8
    int rowBase = tM + ((lane >> 4) << 3);
    #pragma unroll
    for (int t = 0; t < HN_NT; ++t) {
        v8f acc = (t == 0) ? acc0 : (t == 1) ? acc1 : (t == 2) ? acc2 : acc3;
        int col = tN + t * 16 + l15;
        float bv = bias ? bias[col] : 0.0f;
        #pragma unroll
        for (int r = 0; r < 8; ++r) {
            float v = acc[r] + bv;
            if (relu) v = fmaxf(v, 0.0f);
            C[(size_t)(rowBase + r) * ldc + col] = v;
        }
    }
}

// ---------------------------------------------------------------------
// GRU over M=50 steps, C=5 inputs, H=64. One block (64 threads) per
// sequence s (= b*T+t, 512 total). h kept in LDS.
// ---------------------------------------------------------------------
__global__ __launch_bounds__(64) void hn_gru_kernel(
    const float* __restrict__ imu, const float* __restrict__ vel,
    const float* __restrict__ ste,
    const float* __restrict__ w_ih, const float* __restrict__ w_hh,
    const float* __restrict__ b_ih, const float* __restrict__ b_hh,
    float* __restrict__ frame /* (512,64) */)
{
    int s = blockIdx.x;
    int j = threadIdx.x;
    __shared__ float hsh[64];
    hsh[j] = 0.0f;
    __syncthreads();

    const float* imu_s = imu + (size_t)s * 50 * 3;
    const float* vel_s = vel + (size_t)s * 50;
    const float* ste_s = ste + (size_t)s * 50;

    float h = 0.0f;
    for (int mstep = 0; mstep < 50; ++mstep) {
        float xin[5] = { imu_s[mstep * 3 + 0], imu_s[mstep * 3 + 1],
                         imu_s[mstep * 3 + 2], vel_s[mstep], ste_s[mstep] };
        float xr = b_ih[j], xz = b_ih[64 + j], xn = b_ih[128 + j];
        #pragma unroll
        for (int c = 0; c < 5; ++c) {
            xr += w_ih[j * 5 + c]         * xin[c];
            xz += w_ih[(64 + j) * 5 + c]  * xin[c];
            xn += w_ih[(128 + j) * 5 + c] * xin[c];
        }
        float hr = b_hh[j], hz = b_hh[64 + j], hn = b_hh[128 + j];
        for (int c = 0; c < 64; ++c) {
            float hc = hsh[c];
            hr += w_hh[j * 64 + c]         * hc;
            hz += w_hh[(64 + j) * 64 + c]  * hc;
            hn += w_hh[(128 + j) * 64 + c] * hc;
        }
        float r = 1.0f / (1.0f + __expf(-(xr + hr)));
        float z = 1.0f / (1.0f + __expf(-(xz + hz)));
        float n = tanhf(xn + r * hn);
        h = (1.0f - z) * n + z * h;
        __syncthreads();
        hsh[j] = h;
        __syncthreads();
    }
    frame[(size_t)s * 64 + j] = h;
}

// ---------------------------------------------------------------------
// macro_proj (64->128) + reference-exact RoPE.
// rope[t][e] = e<64 ? sin(t*invf[e]) : cos(t*invf[e-64]), invf[k]=10000^(-2k/128)
// sin_used[i]=rope[2i], cos_used[i]=rope[2i+1]; pair (2i,2i+1) from (x1,x2).
// ---------------------------------------------------------------------
__device__ __forceinline__ float hn_ropeval(int t, int e) {
    if (e < 64) {
        float invf = powf(10000.0f, -(2.0f * (float)e) / 128.0f);
        return __sinf((float)t * invf);
    } else {
        float invf = powf(10000.0f, -(2.0f * (float)(e - 64)) / 128.0f);
        return __cosf((float)t * invf);
    }
}

__global__ __launch_bounds__(128) void hn_macro_rope_kernel(
    const float* __restrict__ frame, const float* __restrict__ w,
    const float* __restrict__ b, float* __restrict__ z /* (512,128) */)
{
    int s = blockIdx.x;
    int t = s & 63;              // T = 64
    int j = threadIdx.x;
    __shared__ float row[128];
    const float* f = frame + (size_t)s * 64;
    float acc = b[j];
    for (int c = 0; c < 64; ++c) acc += w[j * 64 + c] * f[c];
    row[j] = acc;
    __syncthreads();
    if (j < 64) {
        float x1 = row[2 * j], x2 = row[2 * j + 1];
        float su = hn_ropeval(t, 2 * j);
        float cu = hn_ropeval(t, 2 * j + 1);
        z[(size_t)s * 128 + 2 * j]     = x1 * cu - x2 * su;
        z[(size_t)s * 128 + 2 * j + 1] = x1 * su + x2 * cu;
    }
}

// ---------------------------------------------------------------------
// Encoder self-attention: T=64 tokens, nh=4, hd=32. One block per (b,h),
// 64 threads = one query each. K/V head staged in LDS.
// ---------------------------------------------------------------------
__global__ __launch_bounds__(64) void hn_enc_attn_kernel(
    const float* __restrict__ qkv /* (512,384) */,
    float* __restrict__ attn_out  /* (512,128) */)
{
    int b = blockIdx.x >> 2, h = blockIdx.x & 3;
    int tq = threadIdx.x;
    __shared__ float Ks[64][33];
    __shared__ float Vs[64][33];
    const float* base = qkv + (size_t)b * 64 * 384;
    #pragma unroll
    for (int d = 0; d < 32; ++d) {
        Ks[tq][d] = base[tq * 384 + 128 + h * 32 + d];
        Vs[tq][d] = base[tq * 384 + 256 + h * 32 + d];
    }
    __syncthreads();
    float q[32];
    #pragma unroll
    for (int d = 0; d < 32; ++d) q[d] = base[tq * 384 + h * 32 + d];

    const float scale = 0.1767766953f;  // 32^-0.5
    float s[64];
    float mx = -1e30f;
    for (int k = 0; k < 64; ++k) {
        float a = 0.0f;
        #pragma unroll
        for (int d = 0; d < 32; ++d) a += q[d] * Ks[k][d];
        s[k] = a * scale;
        mx = fmaxf(mx, s[k]);
    }
    float sum = 0.0f;
    for (int k = 0; k < 64; ++k) { s[k] = __expf(s[k] - mx); sum += s[k]; }
    float inv = 1.0f / sum;
    #pragma unroll
    for (int d = 0; d < 32; ++d) {
        float o = 0.0f;
        for (int k = 0; k < 64; ++k) o += s[k] * Vs[k][d];
        attn_out[((size_t)b * 64 + tq) * 128 + h * 32 + d] = o * inv;
    }
}

// ---------------------------------------------------------------------
// out = optional_relu(LayerNorm(x + y) * w + bias). One block per row,
// blockDim == D (128 or 256). Wave32 shuffle + LDS reduction.
// ---------------------------------------------------------------------
__global__ void hn_add_ln_kernel(
    const float* __restrict__ x, const float* __restrict__ y,
    const float* __restrict__ w, const float* __restrict__ bias,
    float* __restrict__ out, int D, int relu)
{
    int row = blockIdx.x, j = threadIdx.x;
    __shared__ float red[8];
    int nw = blockDim.x >> 5;
    float v = x[(size_t)row * D + j] + (y ? y[(size_t)row * D + j] : 0.0f);

    float sum = v;
    for (int off = 16; off > 0; off >>= 1) sum += __shfl_xor(sum, off, 32);
    if ((j & 31) == 0) red[j >> 5] = sum;
    __syncthreads();
    float tot = 0.0f;
    for (int i = 0; i < nw; ++i) tot += red[i];
    float mean = tot / (float)D;
    __syncthreads();

    float d = v - mean;
    float sq = d * d;
    for (int off = 16; off > 0; off >>= 1) sq += __shfl_xor(sq, off, 32);
    if ((j & 31) == 0) red[j >> 5] = sq;
    __syncthreads();
    float vt = 0.0f;
    for (int i = 0; i < nw; ++i) vt += red[i];
    float var = vt / (float)D;

    float r = d * rsqrtf(var + 1e-5f) * w[j] + bias[j];
    if (relu) r = fmaxf(r, 0.0f);
    out[(size_t)row * D + j] = r;
}

// ---------------------------------------------------------------------
// Row squared-norms (D=128), one wave per row.
// ---------------------------------------------------------------------
__global__ __launch_bounds__(32) void hn_rowsq_kernel(
    const float* __restrict__ xg, float* __restrict__ sq)
{
    int row = blockIdx.x, lane = threadIdx.x;
    float s = 0.0f;
    for (int c = lane; c < 128; c += 32) {
        float v = xg[(size_t)row * 128 + c];
        s += v * v;
    }
    for (int off = 16; off > 0; off >>= 1) s += __shfl_xor(s, off, 32);
    if (lane == 0) sq[row] = s;
}

// ---------------------------------------------------------------------
// kNN top-(K+1) selection per row i of batch b: dist = sqrt(max(0,
// sq_i + sq_j - 2*G[i][j])), invalid -> 1e9. 17 iterative wave-min
// rounds with (value, index) tie-break (matches lax.top_k stability);
// first pick (self) dropped. One wave per row; each lane owns 64 cols.
// ---------------------------------------------------------------------
__global__ __launch_bounds__(32) void hn_topk_kernel(
    const float* __restrict__ G /* (2048,2048) dots for batch b */,
    const float* __restrict__ sq /* (16384) */,
    const unsigned char* __restrict__ mask /* (16384) bool bytes */,
    int b, int* __restrict__ idx /* (8,2048,16) */)
{
    const int N = 2048;
    int i = blockIdx.x;
    int lane = threadIdx.x;
    const float* g = G + (size_t)i * N;
    float sqi = sq[b * N + i];
    bool vi = mask[b * N + i] != 0;

    float dloc[64];
    for (int c = 0; c < 64; ++c) {
        int j = lane + c * 32;
        float d2 = sqi + sq[b * N + j] - 2.0f * g[j];
        float dist = sqrtf(fmaxf(d2, 0.0f));
        bool valid = vi && (mask[b * N + j] != 0);
        dloc[c] = valid ? dist : 1e9f;
    }

    for (int k = 0; k < 17; ++k) {
        float mv = 1e30f;
        int mc = 0;
        for (int c = 0; c < 64; ++c) {
            if (dloc[c] < mv) { mv = dloc[c]; mc = c; }
        }
        int mj = lane + mc * 32;
        for (int off = 16; off > 0; off >>= 1) {
            float ov = __shfl_xor(mv, off, 32);
            int   oj = __shfl_xor(mj, off, 32);
            if (ov < mv || (ov == mv && oj < mj)) { mv = ov; mj = oj; }
        }
        if (k > 0 && lane == 0)
            idx[((size_t)b * 2048 + i) * 16 + (k - 1)] = mj;
        if ((mj & 31) == lane) dloc[mj >> 5] = 1e30f;
    }
}

// ---------------------------------------------------------------------
// GAT neighbor attention: heads=4, dim=32, K=16. One block (128 thr)
// per node bn.
// ---------------------------------------------------------------------
__global__ __launch_bounds__(128) void hn_gat_attn_kernel(
    const float* __restrict__ h /* (16384,128) */,
    const int* __restrict__ idx /* (8,2048,16) */,
    const float* __restrict__ a /* (64) */,
    float* __restrict__ out /* (16384,128) */)
{
    int bn = blockIdx.x;
    int tid = threadIdx.x;
    int b = bn >> 11;
    __shared__ float hc[128];
    __shared__ float alpha[4][16];
    hc[tid] = h[(size_t)bn * 128 + tid];
    __syncthreads();

    const int* myidx = idx + (size_t)bn * 16;
    if (tid < 64) {
        int hH = tid >> 4, k = tid & 15;
        int nb = myidx[k];
        const float* hn = h + ((size_t)(b << 11) + nb) * 128 + hH * 32;
        float acc = 0.0f;
        #pragma unroll
        for (int d = 0; d < 32; ++d)
            acc += a[d] * hc[hH * 32 + d] + a[32 + d] * hn[d];
        alpha[hH][k] = (acc > 0.0f) ? acc : 0.2f * acc;  // leaky relu
    }
    __syncthreads();
    if (tid < 4) {
        float mx = -1e30f;
        for (int k = 0; k < 16; ++k) mx = fmaxf(mx, alpha[tid][k]);
        float s = 0.0f;
        for (int k = 0; k < 16; ++k) {
            float ex = __expf(alpha[tid][k] - mx);
            alpha[tid][k] = ex; s += ex;
        }
        float inv = 1.0f / s;
        for (int k = 0; k < 16; ++k) alpha[tid][k] *= inv;
    }
    __syncthreads();
    int hH = tid >> 5, d = tid & 31;
    float acc = 0.0f;
    for (int k = 0; k < 16; ++k) {
        int nb = myidx[k];
        acc += alpha[hH][k] * h[((size_t)(b << 11) + nb) * 128 + hH * 32 + d];
    }
    out[(size_t)bn * 128 + tid] = acc;   // tid == hH*32+d
}

// ---------------------------------------------------------------------
// Fusion cross-attention: 1 query, 2048 keys, nh=4, hd=64. One block
// (256 threads) per (b,h); scores staged in LDS (8 KB).
// ---------------------------------------------------------------------
__global__ __launch_bounds__(256) void hn_fusion_attn_kernel(
    const float* __restrict__ Q  /* (8,256) */,
    const float* __restrict__ Kp /* (16384,256) */,
    const float* __restrict__ Vp /* (16384,256) */,
    float* __restrict__ o /* (8,256) */)
{
    int b = blockIdx.x >> 2, h = blockIdx.x & 3;
    int tid = threadIdx.x;
    __shared__ float s[2048];
    __shared__ float red[8];
    __shared__ float qs[64];
    if (tid < 64) qs[tid] = Q[b * 256 + h * 64 + tid];
    __syncthreads();

    const float scale = 0.125f;  // 64^-0.5
    float lmax = -1e30f;
    for (int j = tid; j < 2048; j += 256) {
        const float* kr = Kp + ((size_t)b * 2048 + j) * 256 + h * 64;
        float acc = 0.0f;
        for (int d = 0; d < 64; ++d) acc += qs[d] * kr[d];
        acc *= scale;
        s[j] = acc;
        lmax = fmaxf(lmax, acc);
    }
    for (int off = 16; off > 0; off >>= 1) lmax = fmaxf(lmax, __shfl_xor(lmax, off, 32));
    if ((tid & 31) == 0) red[tid >> 5] = lmax;
    __syncthreads();
    float mx = red[0];
    for (int i = 1; i < 8; ++i) mx = fmaxf(mx, red[i]);
    __syncthreads();

    float lsum = 0.0f;
    for (int j = tid; j < 2048; j += 256) {
        float ex = __expf(s[j] - mx);
        s[j] = ex;
        lsum += ex;
    }
    for (int off = 16; off > 0; off >>= 1) lsum += __shfl_xor(lsum, off, 32);
    if ((tid & 31) == 0) red[tid >> 5] = lsum;
    __syncthreads();
    float tot = 0.0f;
    for (int i = 0; i < 8; ++i) tot += red[i];
    float inv = 1.0f / tot;
    __syncthreads();

    if (tid < 64) {
        float acc = 0.0f;
        for (int j = 0; j < 2048; ++j)
            acc += s[j] * Vp[((size_t)b * 2048 + j) * 256 + h * 64 + tid];
        o[b * 256 + h * 64 + tid] = acc * inv;
    }
}

// ---------------------------------------------------------------------
// Small/odd-shape dense: out[r,n] = act(A[r*lda : +K] . W[n] + bias[n]).
// Used for tiny-M heads and the K=3 point-MLP input layer.
// ---------------------------------------------------------------------
__global__ void hn_small_linear_kernel(
    const float* __restrict__ A, const float* __restrict__ W,
    const float* __restrict__ bias, float* __restrict__ out,
    int M, int N, int K, int lda, int relu)
{
    int t = blockIdx.x * blockDim.x + threadIdx.x;
    if (t >= M * N) return;
    int row = t / N, n = t % N;
    const float* a = A + (size_t)row * lda;
    const float* w = W + (size_t)n * K;
    float acc = bias ? bias[n] : 0.0f;
    for (int c = 0; c < K; ++c) acc += a[c] * w[c];
    if (relu) acc = fmaxf(acc, 0.0f);
    out[(size_t)row * N + n] = acc;
}

// =====================================================================
// Host side
// =====================================================================
static inline void launch_gemm(const float* A, const float* W, const float* bias,
                               float* C, int M, int N, int K,
                               int lda, int ldw, int ldc, int relu, hipStream_t st)
{
    // requires: M%16==0, N%(16*HN_NT)==0, K%4==0, lda/ldw even
    long waves = (long)(M / 16) * (long)(N / (16 * HN_NT));
    int blocks = (int)((waves + 7) / 8);
    hn_gemm_wmma_fast<<<blocks, 256, 0, st>>>(A, W, bias, C, M, N, K, lda, ldw, ldc, relu);
}

extern "C" void kernel_launch(void* const* d_in, const int* in_sizes, int n_in,
                              void* d_out, int out_size, void* d_ws, size_t ws_size,
                              hipStream_t stream)
{
    (void)in_sizes; (void)n_in; (void)out_size; (void)ws_size;
    const int B = 8, N = 2048, T = 64;

    // ---- inputs in setup_inputs() insertion order, params flattened in
    //      construction order (gru, macro_proj, enc_layers[2], point_mlp,
    //      gat_layers[3], fusion, head) ----
    const float* pcd  = (const float*)d_in[0];
    const unsigned char* mask = (const unsigned char*)d_in[1];  // jnp.bool_
    const float* imu  = (const float*)d_in[2];
    const float* vel  = (const float*)d_in[3];
    const float* ste  = (const float*)d_in[4];
    int p = 5;
    const float* gru_wih = (const float*)d_in[p++];
    const float* gru_whh = (const float*)d_in[p++];
    const float* gru_bih = (const float*)d_in[p++];
    const float* gru_bhh = (const float*)d_in[p++];
    const float* mp_w = (const float*)d_in[p++];
    const float* mp_b = (const float*)d_in[p++];
    struct EncP { const float *in_w,*in_b,*out_w,*out_b,*l1_w,*l1_b,*l2_w,*l2_b,
                              *ln1_w,*ln1_b,*ln2_w,*ln2_b; } enc[2];
    for (int l = 0; l < 2; ++l) {
        enc[l].in_w  = (const float*)d_in[p++]; enc[l].in_b  = (const float*)d_in[p++];
        enc[l].out_w = (const float*)d_in[p++]; enc[l].out_b = (const float*)d_in[p++];
        enc[l].l1_w  = (const float*)d_in[p++]; enc[l].l1_b  = (const float*)d_in[p++];
        enc[l].l2_w  = (const float*)d_in[p++]; enc[l].l2_b  = (const float*)d_in[p++];
        enc[l].ln1_w = (const float*)d_in[p++]; enc[l].ln1_b = (const float*)d_in[p++];
        enc[l].ln2_w = (const float*)d_in[p++]; enc[l].ln2_b = (const float*)d_in[p++];
    }
    const float* pm_w1 = (const float*)d_in[p++]; const float* pm_b1 = (const float*)d_in[p++];
    const float* pm_w2 = (const float*)d_in[p++]; const float* pm_b2 = (const float*)d_in[p++];
    struct GatP { const float *W,*a,*ln_w,*ln_b; } gat[3];
    for (int l = 0; l < 3; ++l) {
        gat[l].W    = (const float*)d_in[p++]; gat[l].a    = (const float*)d_in[p++];
        gat[l].ln_w = (const float*)d_in[p++]; gat[l].ln_b = (const float*)d_in[p++];
    }
    const float* fu_temp_w = (const float*)d_in[p++]; const float* fu_temp_b = (const float*)d_in[p++];
    const float* fu_spat_w = (const float*)d_in[p++]; const float* fu_spat_b = (const float*)d_in[p++];
    const float* fu_in_w   = (const float*)d_in[p++]; const float* fu_in_b   = (const float*)d_in[p++];
    const float* fu_out_w  = (const float*)d_in[p++]; const float* fu_out_b  = (const float*)d_in[p++];
    const float* fu_ln_w   = (const float*)d_in[p++]; const float* fu_ln_b   = (const float*)d_in[p++];
    const float* hd_w1 = (const float*)d_in[p++]; const float* hd_b1 = (const float*)d_in[p++];
    const float* hd_w2 = (const float*)d_in[p++]; const float* hd_b2 = (const float*)d_in[p++];

    // ---- workspace layout (floats), ~100 MB total ----
    float* ws = (float*)d_ws;
    float* frame   = ws;               ws += (size_t)B * T * 64;
    float* z       = ws;               ws += (size_t)B * T * 128;
    float* tmp     = ws;               ws += (size_t)B * T * 128;
    float* qkv     = ws;               ws += (size_t)B * T * 384;
    float* ff1     = ws;               ws += (size_t)B * T * 2048;
    float* xg      = ws;               ws += (size_t)B * N * 128;
    float* gat_h   = ws;               ws += (size_t)B * N * 128;
    float* gat_out = ws;               ws += (size_t)B * N * 128;
    float* sqbuf   = ws;               ws += (size_t)B * N;
    float* dist    = ws;               ws += (size_t)N * N;      // per-batch reuse
    float* Kspat   = ws;               ws += (size_t)B * N * 256;
    float* Kp      = ws;               ws += (size_t)B * N * 256;
    float* Vp      = ws;               ws += (size_t)B * N * 256;
    float* q0      = ws;               ws += (size_t)B * 256;
    float* Qp      = ws;               ws += (size_t)B * 256;
    float* attn_o  = ws;               ws += (size_t)B * 256;
    float* aproj   = ws;               ws += (size_t)B * 256;
    float* fused   = ws;               ws += (size_t)B * 256;
    float* h1buf   = ws;               ws += (size_t)B * 128;
    int*   idx     = (int*)ws;         // B*N*16 ints

    // ---- 1. GRU over 512 sequences -> frame_feats (512,64) ----
    hn_gru_kernel<<<B * T, 64, 0, stream>>>(imu, vel, ste, gru_wih, gru_whh,
                                            gru_bih, gru_bhh, frame);

    // ---- 2. macro_proj + RoPE -> z (512,128) ----
    hn_macro_rope_kernel<<<B * T, 128, 0, stream>>>(frame, mp_w, mp_b, z);

    // ---- 3. two encoder layers (WMMA GEMMs + attention + LN) ----
    for (int l = 0; l < 2; ++l) {
        launch_gemm(z, enc[l].in_w, enc[l].in_b, qkv, B * T, 384, 128, 128, 128, 384, 0, stream);
        hn_enc_attn_kernel<<<B * 4, 64, 0, stream>>>(qkv, tmp);   // tmp <- attn heads (512,128)
        launch_gemm(tmp, enc[l].out_w, enc[l].out_b, ff1 /*scratch*/, B * T, 128, 128, 128, 128, 128, 0, stream);
        hn_add_ln_kernel<<<B * T, 128, 0, stream>>>(ff1, z, enc[l].ln1_w, enc[l].ln1_b, z, 128, 0);
        launch_gemm(z, enc[l].l1_w, enc[l].l1_b, ff1, B * T, 2048, 128, 128, 128, 2048, 1, stream);
        launch_gemm(ff1, enc[l].l2_w, enc[l].l2_b, tmp, B * T, 128, 2048, 2048, 2048, 128, 0, stream);
        hn_add_ln_kernel<<<B * T, 128, 0, stream>>>(tmp, z, enc[l].ln2_w, enc[l].ln2_b, z, 128, 0);
    }

    // ---- 4. point MLP: pcd (16384,3) -> 128 (scalar, K=3) -> 128 (WMMA) ----
    hn_small_linear_kernel<<<(B * N * 128 + 255) / 256, 256, 0, stream>>>(
        pcd, pm_w1, pm_b1, gat_h /*scratch*/, B * N, 128, 3, 3, 1);
    launch_gemm(gat_h, pm_w2, pm_b2, xg, B * N, 128, 128, 128, 128, 128, 1, stream);

    // ---- 5. three GAT layers ----
    for (int l = 0; l < 3; ++l) {
        hn_rowsq_kernel<<<B * N, 32, 0, stream>>>(xg, sqbuf);
        for (int b = 0; b < B; ++b) {
            const float* xb = xg + (size_t)b * N * 128;
            // x @ x^T via WMMA (the dominant GEMM, 2048x2048x128 per batch)
            launch_gemm(xb, xb, nullptr, dist, N, N, 128, 128, 128, N, 0, stream);
            hn_topk_kernel<<<N, 32, 0, stream>>>(dist, sqbuf, mask, b, idx);
        }
        launch_gemm(xg, gat[l].W, nullptr, gat_h, B * N, 128, 128, 128, 128, 128, 0, stream);
        hn_gat_attn_kernel<<<B * N, 128, 0, stream>>>(gat_h, idx, gat[l].a, gat_out);
        hn_add_ln_kernel<<<B * N, 128, 0, stream>>>(gat_out, xg, gat[l].ln_w, gat[l].ln_b, xg, 128, 1);
    }

    // ---- 6. fusion ----
    // temporal_feat = z[:, T-1, :]  (rows strided by T*128)
    hn_small_linear_kernel<<<(B * 256 + 255) / 256, 256, 0, stream>>>(
        z + (size_t)(T - 1) * 128, fu_temp_w, fu_temp_b, q0, B, 256, 128, T * 128, 0);
    launch_gemm(xg, fu_spat_w, fu_spat_b, Kspat, B * N, 256, 128, 128, 128, 256, 0, stream);
    // MHA projections (in_w rows: q 0..255, k 256..511, v 512..767)
    hn_small_linear_kernel<<<(B * 256 + 255) / 256, 256, 0, stream>>>(
        q0, fu_in_w, fu_in_b, Qp, B, 256, 256, 256, 0);
    launch_gemm(Kspat, fu_in_w + 256 * 256, fu_in_b + 256, Kp, B * N, 256, 256, 256, 256, 256, 0, stream);
    launch_gemm(Kspat, fu_in_w + 512 * 256, fu_in_b + 512, Vp, B * N, 256, 256, 256, 256, 256, 0, stream);
    hn_fusion_attn_kernel<<<B * 4, 256, 0, stream>>>(Qp, Kp, Vp, attn_o);
    hn_small_linear_kernel<<<(B * 256 + 255) / 256, 256, 0, stream>>>(
        attn_o, fu_out_w, fu_out_b, aproj, B, 256, 256, 256, 0);
    hn_add_ln_kernel<<<B, 256, 0, stream>>>(aproj, q0, fu_ln_w, fu_ln_b, fused, 256, 0);

    // ---- 7. head -> d_out (8,2) ----
    hn_small_linear_kernel<<<(B * 128 + 255) / 256, 256, 0, stream>>>(
        fused, hd_w1, hd_b1, h1buf, B, 128, 256, 256, 1);
    hn_small_linear_kernel<<<1, 64, 0, stream>>>(
        h1buf, hd_w2, hd_b2, (float*)d_out, B, 2, 128, 128, 0);
}